// ManifoldNetSPD_30683246362856
// MI455X (gfx1250) — compile-verified
//
#include <hip/hip_runtime.h>
#include <math.h>

typedef __attribute__((ext_vector_type(2))) float v2f;
typedef __attribute__((ext_vector_type(8))) float v8f;

#define NB   8          // batch
#define NC   6          // unique symmetric components (a00,a01,a02,a11,a12,a22)
#define B6   48         // NB * NC  -> conv batch dimension
#define LDSW 104        // LDS row stride (floats): >= 96+KK-1, constant-folds ds offsets

// ------------------------------------------------------------------
// gfx1250 async global->LDS copy (ASYNCcnt-tracked), with sync fallback
// ------------------------------------------------------------------
#if defined(__has_builtin)
#if __has_builtin(__builtin_amdgcn_global_load_async_to_lds_b32) && \
    __has_builtin(__builtin_amdgcn_s_wait_asynccnt)
#define HAVE_ASYNC_LDS 1
#endif
#endif

__device__ __forceinline__ void ld_async_f32(const float* g, float* l) {
#ifdef HAVE_ASYNC_LDS
  // generic->AS pointers via integer detour (HW truncates LDS generic addr to
  // its low-32-bit offset; global generic == AS1 numerically).
  __builtin_amdgcn_global_load_async_to_lds_b32(
      (__attribute__((address_space(1))) int*)(unsigned long long)g,
      (__attribute__((address_space(3))) int*)(unsigned int)(unsigned long long)l,
      0, 0);
#else
  *l = *g;
#endif
}

__device__ __forceinline__ void wait_async_lds() {
#ifdef HAVE_ASYNC_LDS
  __builtin_amdgcn_s_wait_asynccnt(0);
#endif
}

// ------------------------------------------------------------------
// Analytic eigenvalues of a symmetric 3x3 (Smith's trigonometric method)
// ------------------------------------------------------------------
__device__ __forceinline__ void eig3(float a00, float a01, float a02,
                                     float a11, float a12, float a22,
                                     float& e0, float& e1, float& e2) {
  float q   = (a00 + a11 + a22) * (1.0f / 3.0f);
  float b00 = a00 - q, b11 = a11 - q, b22 = a22 - q;
  float p2  = b00 * b00 + b11 * b11 + b22 * b22 +
              2.0f * (a01 * a01 + a02 * a02 + a12 * a12);
  float p  = sqrtf(p2 * (1.0f / 6.0f)) + 1e-30f;
  float ip = 1.0f / p;
  float c00 = b00 * ip, c01 = a01 * ip, c02 = a02 * ip;
  float c11 = b11 * ip, c12 = a12 * ip, c22 = b22 * ip;
  float detB = c00 * (c11 * c22 - c12 * c12)
             - c01 * (c01 * c22 - c12 * c02)
             + c02 * (c01 * c12 - c11 * c02);
  float r = fminf(1.0f, fmaxf(-1.0f, 0.5f * detB));
  float phi = acosf(r) * (1.0f / 3.0f);
  float t2 = q + 2.0f * p * __cosf(phi);
  float t0 = q + 2.0f * p * __cosf(phi + 2.0943951023931953f); // +2*pi/3
  e0 = t0; e2 = t2; e1 = 3.0f * q - t0 - t2;
}

// f(A) = f(e0)*I + d01*(A - e0 I) + d012*(A - e0 I)(A - e1 I)
#define MATFUN_BODY(FUN)                                                      \
  float e0, e1, e2;                                                           \
  eig3(a00, a01, a02, a11, a12, a22, e0, e1, e2);                             \
  float f0 = FUN(e0), f1 = FUN(e1), f2 = FUN(e2);                             \
  float h10 = e1 - e0, h21 = e2 - e1, h20 = e2 - e0;                          \
  h10 += (h10 >= 0.f ? 1e-20f : -1e-20f);                                     \
  h21 += (h21 >= 0.f ? 1e-20f : -1e-20f);                                     \
  h20 += (h20 >= 0.f ? 1e-20f : -1e-20f);                                     \
  float d01 = (f1 - f0) / h10;                                                \
  float d12 = (f2 - f1) / h21;                                                \
  float d2  = (d12 - d01) / h20;                                              \
  float s00 = a00 * a00 + a01 * a01 + a02 * a02;                              \
  float s01 = a00 * a01 + a01 * a11 + a02 * a12;                              \
  float s02 = a00 * a02 + a01 * a12 + a02 * a22;                              \
  float s11 = a01 * a01 + a11 * a11 + a12 * a12;                              \
  float s12 = a01 * a02 + a11 * a12 + a12 * a22;                              \
  float s22 = a02 * a02 + a12 * a12 + a22 * a22;                              \
  float uv = e0 * e1, upv = e0 + e1;                                          \
  r00 = f0 + d01 * (a00 - e0) + d2 * (s00 - upv * a00 + uv);                  \
  r11 = f0 + d01 * (a11 - e0) + d2 * (s11 - upv * a11 + uv);                  \
  r22 = f0 + d01 * (a22 - e0) + d2 * (s22 - upv * a22 + uv);                  \
  r01 = d01 * a01 + d2 * (s01 - upv * a01);                                   \
  r02 = d01 * a02 + d2 * (s02 - upv * a02);                                   \
  r12 = d01 * a12 + d2 * (s12 - upv * a12);

__device__ __forceinline__ float dev_logf(float x) { return __logf(fmaxf(x, 1e-30f)); }
__device__ __forceinline__ float dev_expf(float x) { return __expf(x); }

// ------------------------------------------------------------------
// logm: x [8,1,96,96,3,3] -> L planes [(b*6+c), 96,96]
// ------------------------------------------------------------------
__global__ void logm_kernel(const float* __restrict__ x, float* __restrict__ L) {
  int idx = blockIdx.x * blockDim.x + threadIdx.x;
  const int total = NB * 96 * 96;
  if (idx >= total) return;
  const float* m = x + (size_t)idx * 9;
  float a00 = m[0], a01 = m[1], a02 = m[2], a11 = m[4], a12 = m[5], a22 = m[8];
  float r00, r01, r02, r11, r12, r22;
  MATFUN_BODY(dev_logf)
  int b = idx / (96 * 96);
  int pix = idx - b * 96 * 96;
  float comps[NC] = {r00, r01, r02, r11, r12, r22};
#pragma unroll
  for (int c = 0; c < NC; ++c)
    L[(size_t)(b * NC + c) * (96 * 96) + pix] = comps[c];
}

// ------------------------------------------------------------------
// expm: S planes [(b*6+c),96,96] -> out [8,1,96,96,3,3]
// ------------------------------------------------------------------
__global__ void expm_kernel(const float* __restrict__ S, float* __restrict__ out) {
  int idx = blockIdx.x * blockDim.x + threadIdx.x;
  const int total = NB * 96 * 96;
  if (idx >= total) return;
  int b = idx / (96 * 96);
  int pix = idx - b * 96 * 96;
  const float* base = S + (size_t)(b * NC) * (96 * 96) + pix;
  float a00 = base[0 * 96 * 96], a01 = base[1 * 96 * 96], a02 = base[2 * 96 * 96];
  float a11 = base[3 * 96 * 96], a12 = base[4 * 96 * 96], a22 = base[5 * 96 * 96];
  float r00, r01, r02, r11, r12, r22;
  MATFUN_BODY(dev_expf)
  float* o = out + (size_t)idx * 9;
  o[0] = r00; o[1] = r01; o[2] = r02;
  o[3] = r01; o[4] = r11; o[5] = r12;
  o[6] = r02; o[7] = r12; o[8] = r22;
}

// ------------------------------------------------------------------
// softmax of conv weights, one row per thread; rows padded to Kt4 with zeros
// ------------------------------------------------------------------
__global__ void softmax_rows(const float* __restrict__ w, float* __restrict__ wn,
                             int Co, int Kt, int Kt4) {
  int row = blockIdx.x * blockDim.x + threadIdx.x;
  if (row >= Co) return;
  const float* wr = w + (size_t)row * Kt;
  float mx = -1e30f;
  for (int i = 0; i < Kt; ++i) mx = fmaxf(mx, wr[i]);
  float s = 0.f;
  for (int i = 0; i < Kt; ++i) s += __expf(wr[i] - mx);
  float inv = 1.0f / s;
  float* o = wn + (size_t)row * Kt4;
  for (int i = 0; i < Kt; ++i) o[i] = __expf(wr[i] - mx) * inv;
  for (int i = Kt; i < Kt4; ++i) o[i] = 0.f;
}

// ------------------------------------------------------------------
// Implicit-GEMM conv via V_WMMA_F32_16X16X4_F32, LDS-staged inputs.
//   One block (192 thr = 6 waves) per (b6, y) output row (Wo <= 96 = 6 tiles).
//   Input strip CI x KK rows x CW cols staged once via async global->LDS
//   (zero-pad of the tangent space folded into the staging predicate),
//   then the whole K loop reads B fragments from LDS at constant offsets.
// ------------------------------------------------------------------
template <int CI, int CO, int KK, int PAD>
__global__ void __launch_bounds__(192)
spd_conv_wmma(const float* __restrict__ in, const float* __restrict__ wn,
              float* __restrict__ out, int HiBuf, int WiBuf, int Ho, int Wo) {
  constexpr int KT  = CI * KK * KK;
  constexpr int KT4 = (KT + 3) & ~3;
  constexpr int RWS = CI * KK;            // staged input rows
  __shared__ float sw[CO * KT4];
  __shared__ float stile[RWS * LDSW];

  const int tid  = threadIdx.x;
  const int y    = blockIdx.x % Ho;
  const int b6   = blockIdx.x / Ho;
  const int Hv   = HiBuf - PAD, Wv = WiBuf - PAD;
  const int CW   = Wo + KK - 1;           // columns actually needed
  const float* inb = in + (size_t)b6 * CI * HiBuf * WiBuf;

  // stage softmaxed weights
  for (int i = tid; i < CO * KT4; i += 192) sw[i] = wn[i];

  // stage input strip: async copy valid elements, zero the rest
  for (int i = tid; i < RWS * LDSW; i += 192) {
    const int r  = i / LDSW;              // r = ci*KK + ky
    const int cx = i - r * LDSW;
    const int ci = r / KK;
    const int ky = r - ci * KK;
    const int sy = y + ky - PAD;
    const int sx = cx - PAD;
    const bool ok = (cx < CW) & (sy >= 0) & (sy < Hv) & (sx >= 0) & (sx < Wv);
    if (ok)
      ld_async_f32(&inb[(size_t)(ci * HiBuf + sy) * WiBuf + sx], &stile[i]);
    else
      stile[i] = 0.f;
  }
  wait_async_lds();
  __syncthreads();

  const int lane = tid & 31;
  const int wave = tid >> 5;              // 0..5, one 16-wide tile each
  const int x0   = wave << 4;
  if (x0 >= Wo) return;                   // wave-uniform: EXEC all-1s below

  const int n    = lane & 15;             // A row index == B column index
  const int hi   = lane >> 4;             // K half per f32 WMMA layout
  const int cbase = x0 + n;               // lane's column in the staged strip

  v8f acc = {};
#pragma unroll
  for (int kk = 0; kk < KT4; kk += 4) {
    const int ka = kk + (hi << 1);
    v2f a, b;
    a.x = (n < CO) ? sw[n * KT4 + ka] : 0.f;
    a.y = (n < CO) ? sw[n * KT4 + ka + 1] : 0.f;
#pragma unroll
    for (int e = 0; e < 2; ++e) {
      const int kidx = ka + e;
      float v = 0.f;
      if (kidx < KT) {
        const int r  = kidx / KK;         // = ci*KK + ky
        const int kx = kidx - r * KK;
        v = stile[r * LDSW + cbase + kx];
      }
      if (e == 0) b.x = v; else b.y = v;
    }
    acc = __builtin_amdgcn_wmma_f32_16x16x4_f32(
        false, a, false, b, (short)0, acc, false, false);
  }

  const int x = x0 + n;
  if (x < Wo) {
#pragma unroll
    for (int r = 0; r < 8; ++r) {
      const int m = (hi << 3) + r;
      if (m < CO)
        out[(((size_t)b6 * CO + m) * Ho + y) * Wo + x] = acc[r];
    }
  }
}

// ------------------------------------------------------------------
extern "C" void kernel_launch(void* const* d_in, const int* in_sizes, int n_in,
                              void* d_out, int out_size, void* d_ws, size_t ws_size,
                              hipStream_t stream) {
  (void)in_sizes; (void)n_in; (void)out_size; (void)ws_size;
  const float* x = (const float*)d_in[0];
  const float* w[10];
  for (int i = 0; i < 10; ++i) w[i] = (const float*)d_in[1 + i];

  static const int Co_[10] = {4, 8, 16, 16, 8, 16, 16, 8, 4, 1};
  static const int Ci_[10] = {1, 4, 8, 16, 16, 8, 16, 16, 8, 4};
  static const int k_[10]  = {3, 3, 3, 2, 2, 2, 2, 3, 3, 3};

  float* ws = (float*)d_ws;
  float* wn[10];
  size_t off = 0;
  int Kt[10], Kt4[10];
  for (int i = 0; i < 10; ++i) {
    Kt[i]  = Ci_[i] * k_[i] * k_[i];
    Kt4[i] = (Kt[i] + 3) & ~3;
    wn[i]  = ws + off;
    off += (size_t)Co_[i] * Kt4[i];
  }
  float* bufA = ws + 8192;
  float* bufB = bufA + 6500000;   // max tangent tensor: 48*16*90*90 = 6.22M floats

  for (int i = 0; i < 10; ++i)
    softmax_rows<<<1, 32, 0, stream>>>(w[i], wn[i], Co_[i], Kt[i], Kt4[i]);

  const int npix = NB * 96 * 96;
  logm_kernel<<<(npix + 255) / 256, 256, 0, stream>>>(x, bufA);

#define CONV(L, CI, CO, KK, PAD, SRC, DST, HiB, WiB, Ho, Wo)                    \
  do {                                                                          \
    spd_conv_wmma<CI, CO, KK, PAD>                                              \
        <<<B6 * (Ho), 192, 0, stream>>>(SRC, wn[L], DST, HiB, WiB, Ho, Wo);     \
  } while (0)

  CONV(0, 1, 4, 3, 0, bufA, bufB, 96, 96, 94, 94);
  CONV(1, 4, 8, 3, 0, bufB, bufA, 94, 94, 92, 92);
  CONV(2, 8, 16, 3, 0, bufA, bufB, 92, 92, 90, 90);
  CONV(3, 16, 16, 2, 0, bufB, bufA, 90, 90, 89, 89);
  CONV(4, 16, 8, 2, 0, bufA, bufB, 89, 89, 88, 88);
  CONV(5, 8, 16, 2, 2, bufB, bufA, 88, 88, 89, 89);   // logical 90 after pad 2
  CONV(6, 16, 16, 2, 2, bufA, bufB, 89, 89, 90, 90);  // logical 91 after pad 2
  CONV(7, 16, 8, 3, 4, bufB, bufA, 90, 90, 92, 92);   // logical 94 after pad 4
  CONV(8, 8, 4, 3, 4, bufA, bufB, 92, 92, 94, 94);    // logical 96 after pad 4
  CONV(9, 4, 1, 3, 4, bufB, bufA, 94, 94, 96, 96);    // logical 98 after pad 4
#undef CONV

  expm_kernel<<<(npix + 255) / 256, 256, 0, stream>>>(bufA, (float*)d_out);
}